// GCN_56384330662076
// MI455X (gfx1250) — compile-verified
//
#include <hip/hip_runtime.h>
#include <hip/hip_bf16.h>

typedef __attribute__((ext_vector_type(2))) float v2f;
typedef __attribute__((ext_vector_type(8))) float v8f;

#define F1 128
#define F2 64

// ---------------- degree / norm precompute ----------------

__global__ void k_init_deg(float* __restrict__ deg, int n) {
    int i = blockIdx.x * blockDim.x + threadIdx.x;
    if (i < n) deg[i] = 1.0f;  // self-loop contributes weight 1
}

__global__ void k_accum_deg(const int* __restrict__ col, const float* __restrict__ ew,
                            float* __restrict__ deg, int e) {
    int i = blockIdx.x * blockDim.x + threadIdx.x;
    if (i < e) atomicAdd(&deg[col[i]], ew[i]);
}

__global__ void k_rsqrt_inplace(float* __restrict__ d, int n) {
    int i = blockIdx.x * blockDim.x + threadIdx.x;
    if (i < n) {
        float v = d[i];
        d[i] = (v > 0.0f) ? rsqrtf(v) : 0.0f;
    }
}

__global__ void k_edge_norm(const int* __restrict__ row, const int* __restrict__ col,
                            const float* __restrict__ ew, const float* __restrict__ dis,
                            float* __restrict__ norm, int e) {
    int i = blockIdx.x * blockDim.x + threadIdx.x;
    if (i < e) norm[i] = dis[row[i]] * ew[i] * dis[col[i]];
}

// ---------------- fp32 WMMA GEMM: C[M,N] = A[M,K] * B[K,N] ----------------
// grid.x = ceil(M/16); blockDim.x = (N/16) * 32  (one 16x16 N-tile per wave)
// Uses V_WMMA_F32_16X16X4_F32: A 16x4, B 4x16, C 16x16 f32 (ISA 7.12.2 layouts):
//   A: lane%16 = M row; VGPRs {0,1} = K { 2*(lane/16), 2*(lane/16)+1 }
//   B: lane%16 = N col; VGPRs {0,1} = K { 2*(lane/16), 2*(lane/16)+1 }
//   C: lane%16 = N col; VGPR r = M row (r + 8*(lane/16))
__global__ void k_gemm_wmma_f32(const float* __restrict__ A, const float* __restrict__ B,
                                float* __restrict__ C, int M, int K, int N) {
    const int lane = threadIdx.x & 31;
    const int wave = threadIdx.x >> 5;
    const int mt   = blockIdx.x;        // M tile
    const int nt   = wave;              // N tile
    const int half = lane >> 4;         // 0: lanes 0-15, 1: lanes 16-31
    const int l16  = lane & 15;

    int mrow = mt * 16 + l16;
    if (mrow > M - 1) mrow = M - 1;     // clamp loads; stores guarded below
    const int ncol = nt * 16 + l16;

    const float* __restrict__ Arow = A + (size_t)mrow * K;

    v8f acc = {};
    for (int k0 = 0; k0 < K; k0 += 4) {
        const int ka = k0 + 2 * half;
        v2f a, b;
        a.x = Arow[ka];
        a.y = Arow[ka + 1];
        b.x = B[(size_t)ka * N + ncol];
        b.y = B[(size_t)(ka + 1) * N + ncol];
        acc = __builtin_amdgcn_wmma_f32_16x16x4_f32(
            /*neg_a=*/false, a, /*neg_b=*/false, b,
            /*c_mod=*/(short)0, acc, /*reuse_a=*/false, /*reuse_b=*/false);
    }

    const int mbase = mt * 16 + 8 * half;
#pragma unroll
    for (int r = 0; r < 8; ++r) {
        int m = mbase + r;
        if (m < M) C[(size_t)m * N + ncol] = acc[r];
    }
}

// ---------------- aggregation ----------------

// agg[i][f] = dis[i]^2 * xw[i][f] + bias[f]   (self-loop term + bias)
__global__ void k_init_agg(const float* __restrict__ xw, const float* __restrict__ dis,
                           const float* __restrict__ bias, float* __restrict__ agg,
                           int total, int logF) {
    int i = blockIdx.x * blockDim.x + threadIdx.x;
    if (i < total) {
        int node = i >> logF;
        int f = i & ((1 << logF) - 1);
        float d = dis[node];
        agg[i] = d * d * xw[i] + bias[f];
    }
}

// one wave per edge, 4 floats/lane (128 features)
__global__ void k_scatter128(const float* __restrict__ src, const int* __restrict__ row,
                             const int* __restrict__ col, const float* __restrict__ norm,
                             float* __restrict__ dst, int e) {
    int warp = (blockIdx.x * blockDim.x + threadIdx.x) >> 5;
    int lane = threadIdx.x & 31;
    if (warp >= e) return;
    const int r = row[warp];
    const int c = col[warp];
    const float w = norm[warp];
    float4 v = ((const float4*)(src + (size_t)r * F1))[lane];
    float* d = dst + (size_t)c * F1 + lane * 4;
    atomicAdd(d + 0, v.x * w);
    atomicAdd(d + 1, v.y * w);
    atomicAdd(d + 2, v.z * w);
    atomicAdd(d + 3, v.w * w);
}

// one wave per edge, 2 floats/lane (64 features)
__global__ void k_scatter64(const float* __restrict__ src, const int* __restrict__ row,
                            const int* __restrict__ col, const float* __restrict__ norm,
                            float* __restrict__ dst, int e) {
    int warp = (blockIdx.x * blockDim.x + threadIdx.x) >> 5;
    int lane = threadIdx.x & 31;
    if (warp >= e) return;
    const int r = row[warp];
    const int c = col[warp];
    const float w = norm[warp];
    float2 v = ((const float2*)(src + (size_t)r * F2))[lane];
    float* d = dst + (size_t)c * F2 + lane * 2;
    atomicAdd(d + 0, v.x * w);
    atomicAdd(d + 1, v.y * w);
}

__global__ void k_relu_inplace(float* __restrict__ h, int total) {
    int i = blockIdx.x * blockDim.x + threadIdx.x;
    if (i < total) h[i] = fmaxf(h[i], 0.0f);
}

// one wave per node row of 64, 2 values/lane; wave32 shuffle reductions
__global__ void k_log_softmax64(float* __restrict__ out, int n) {
    int warp = (blockIdx.x * blockDim.x + threadIdx.x) >> 5;
    int lane = threadIdx.x & 31;
    if (warp >= n) return;
    float2* rowp = (float2*)(out + (size_t)warp * F2);
    float2 v = rowp[lane];
    float m = fmaxf(v.x, v.y);
#pragma unroll
    for (int off = 16; off > 0; off >>= 1) m = fmaxf(m, __shfl_xor(m, off, 32));
    float s = expf(v.x - m) + expf(v.y - m);
#pragma unroll
    for (int off = 16; off > 0; off >>= 1) s += __shfl_xor(s, off, 32);
    float lse = m + logf(s);
    rowp[lane] = make_float2(v.x - lse, v.y - lse);
}

// ---------------- launch ----------------

extern "C" void kernel_launch(void* const* d_in, const int* in_sizes, int n_in,
                              void* d_out, int out_size, void* d_ws, size_t ws_size,
                              hipStream_t stream) {
    const float* x  = (const float*)d_in[0];
    const int*   ei = (const int*)d_in[1];
    const float* ew = (const float*)d_in[2];
    const float* W1 = (const float*)d_in[3];
    const float* b1 = (const float*)d_in[4];
    const float* W2 = (const float*)d_in[5];
    const float* b2 = (const float*)d_in[6];
    float* out = (float*)d_out;

    const int N = in_sizes[0] / F1;   // 50000
    const int E = in_sizes[1] / 2;    // 800000
    const int* row = ei;
    const int* col = ei + E;

    // workspace layout (floats)
    float* dis  = (float*)d_ws;                  // N     (deg -> rsqrt in place)
    float* norm = dis + N;                       // E
    float* xw   = norm + E;                      // N*F1
    float* agg1 = xw + (size_t)N * F1;           // N*F1  (-> relu h in place)
    float* hw   = agg1 + (size_t)N * F1;         // N*F2

    const int TB = 256;
    const int gN   = (N + TB - 1) / TB;
    const int gE   = (E + TB - 1) / TB;
    const int gNF1 = (N * F1 + TB - 1) / TB;
    const int warpsPerBlock = TB / 32;
    const int gEw = (E + warpsPerBlock - 1) / warpsPerBlock;   // warp-per-edge grids
    const int gNw = (N + warpsPerBlock - 1) / warpsPerBlock;   // warp-per-node grids
    const int gM16 = (N + 15) / 16;                            // 3125 M-tiles

    // 1) symmetric normalization coefficients
    k_init_deg<<<gN, TB, 0, stream>>>(dis, N);
    k_accum_deg<<<gE, TB, 0, stream>>>(col, ew, dis, E);
    k_rsqrt_inplace<<<gN, TB, 0, stream>>>(dis, N);
    k_edge_norm<<<gE, TB, 0, stream>>>(row, col, ew, dis, norm, E);

    // 2) layer 1: xw = x @ W1   (WMMA f32, 8 N-tiles -> 256 threads/block)
    k_gemm_wmma_f32<<<gM16, (F1 / 16) * 32, 0, stream>>>(x, W1, xw, N, F1, F1);

    // 3) agg1 = dis^2*xw + b1, scatter edges, relu
    k_init_agg<<<gNF1, TB, 0, stream>>>(xw, dis, b1, agg1, N * F1, 7);
    k_scatter128<<<gEw, TB, 0, stream>>>(xw, row, col, norm, agg1, E);
    k_relu_inplace<<<gNF1, TB, 0, stream>>>(agg1, N * F1);

    // 4) layer 2: hw = h @ W2   (4 N-tiles -> 128 threads/block)
    k_gemm_wmma_f32<<<gM16, (F2 / 16) * 32, 0, stream>>>(agg1, W2, hw, N, F1, F2);

    // 5) out = dis^2*hw + b2, scatter edges, log_softmax in place
    k_init_agg<<<(N * F2 + TB - 1) / TB, TB, 0, stream>>>(hw, dis, b2, out, N * F2, 6);
    k_scatter64<<<gEw, TB, 0, stream>>>(hw, row, col, norm, out, E);
    k_log_softmax64<<<gNw, TB, 0, stream>>>(out, N);
}